// MultiHeadAttention_53420803227717
// MI455X (gfx1250) — compile-verified
//
#include <hip/hip_runtime.h>
#include <hip/hip_bf16.h>
#include <math.h>

// MHA forward for MI455X (gfx1250, wave32).
// fp32 tensor path V_WMMA_F32_16X16X4_F32 everywhere (problem is bound by the
// 256MB attn_weights write; fp32 WMMA matches the fp32 reference exactly).
//
// d_out layout (tuple return): [ out: 4M floats | attn_weights: 64M floats ]
// d_ws layout: [ Q | K | V | ctx ] each 4M floats = 64MB total.

#define B_ 4
#define S_ 1024
#define D_ 1024
#define H_ 16
#define DK_ 64

// LDS row stride for 16-row strips: 1028 ≡ 4 (mod 64 banks) makes the
// 16-rows-at-one-column-pair fragment reads conflict-free for both lane
// halves, and keeps rows 16B-aligned for b128 LDS ops.
#define A_STRIDE 1028

typedef float v2f __attribute__((ext_vector_type(2)));
typedef float v8f __attribute__((ext_vector_type(8)));

// Fragment mapping for V_WMMA_F32_16X16X4_F32 (CDNA5 ISA 7.12.2, wave32):
//   A (16x4):  lane l -> M = l%16, a.x = A[M, 2*(l/16)],   a.y = A[M, 2*(l/16)+1]
//   B (4x16):  lane l -> N = l%16, b.x = B[2*(l/16), N],   b.y = B[2*(l/16)+1, N]
//   C/D(16x16):lane l -> N = l%16, vgpr v -> M = v + 8*(l/16)

__device__ __forceinline__ v8f wmma_f32(v2f a, v2f b, v8f c) {
    return __builtin_amdgcn_wmma_f32_16x16x4_f32(
        false, a, false, b, (short)0, c, false, false);
}

// ---------------------------------------------------------------------------
// Kernel 1: QKV projection. y[r,n] = sum_k x[r,k]*W[n,k] + b[n]
// One block per (matrix via blockIdx.y, row-strip of 16). A-strip staged in
// LDS once; each of 8 waves computes two 16x64 slabs (4 wmma per A-fragment).
// Output head-split [B,H,S,dk].
// ---------------------------------------------------------------------------
__global__ void __launch_bounds__(256)
mha_qkv_proj(const float* __restrict__ qin, const float* __restrict__ kin,
             const float* __restrict__ vin,
             const float* __restrict__ Wq, const float* __restrict__ Wk,
             const float* __restrict__ Wv,
             const float* __restrict__ bq, const float* __restrict__ bk,
             const float* __restrict__ bv,
             float* __restrict__ Qo, float* __restrict__ Ko,
             float* __restrict__ Vo)
{
    extern __shared__ float lds[];            // [16][A_STRIDE]

    const float* x; const float* W; const float* bias; float* out;
    if (blockIdx.y == 0)      { x = qin; W = Wq; bias = bq; out = Qo; }
    else if (blockIdx.y == 1) { x = kin; W = Wk; bias = bk; out = Ko; }
    else                      { x = vin; W = Wv; bias = bv; out = Vo; }

    const int mt = blockIdx.x;                // 0..255 row strip
    const int m0 = mt << 4;

    // ---- cooperative A-strip load: 16 rows x 1024 floats -> LDS ----
    {
        const float4* src = (const float4*)(x + (size_t)m0 * D_);
        for (int f = threadIdx.x; f < 16 * (D_ / 4); f += 256) {
            const int row = f >> 8;           // 256 float4 per row
            const int c4  = f & 255;
            *(float4*)&lds[row * A_STRIDE + c4 * 4] = src[row * 256 + c4];
        }
    }
    __syncthreads();

    const int wave = threadIdx.x >> 5;
    const int lane = threadIdx.x & 31;
    const int lh   = lane >> 4;
    const int lm   = lane & 15;
    const float* aL = lds + lm * A_STRIDE;

    for (int p = 0; p < 2; ++p) {
        const int ng = p * 8 + wave;          // 0..15 group of 64 cols
        const int n0 = ng << 6;
        const float* bRow0 = W + (size_t)(n0 + lm) * D_;
        const float* bRow1 = bRow0 + (size_t)16 * D_;
        const float* bRow2 = bRow0 + (size_t)32 * D_;
        const float* bRow3 = bRow0 + (size_t)48 * D_;
        __builtin_prefetch(bRow0, 0, 0);
        __builtin_prefetch(bRow1, 0, 0);
        __builtin_prefetch(bRow2, 0, 0);
        __builtin_prefetch(bRow3, 0, 0);

        v8f acc[4];
        acc[0] = acc[1] = acc[2] = acc[3] = (v8f){0.f,0.f,0.f,0.f,0.f,0.f,0.f,0.f};

#pragma unroll 4
        for (int kk = 0; kk < D_; kk += 4) {
            const int ks = kk + lh * 2;
            v2f a;  a.x  = aL[ks];      a.y  = aL[ks + 1];
            v2f b0; b0.x = bRow0[ks];   b0.y = bRow0[ks + 1];
            v2f b1; b1.x = bRow1[ks];   b1.y = bRow1[ks + 1];
            v2f b2; b2.x = bRow2[ks];   b2.y = bRow2[ks + 1];
            v2f b3; b3.x = bRow3[ks];   b3.y = bRow3[ks + 1];
            acc[0] = wmma_f32(a, b0, acc[0]);
            acc[1] = wmma_f32(a, b1, acc[1]);
            acc[2] = wmma_f32(a, b2, acc[2]);
            acc[3] = wmma_f32(a, b3, acc[3]);
        }

        const int h = n0 >> 6;                // whole 64-col group = one head
#pragma unroll
        for (int j = 0; j < 4; ++j) {
            const float bval = bias[n0 + j * 16 + lm];
#pragma unroll
            for (int vv = 0; vv < 8; ++vv) {
                const int r  = m0 + vv + 8 * lh;
                const int bb = r >> 10;
                const int s  = r & 1023;
                out[(((size_t)(bb * H_ + h) * S_) + s) * DK_ + j * 16 + lm] =
                    acc[j][vv] + bval;
            }
        }
    }
}

// ---------------------------------------------------------------------------
// Kernel 2: attention core. One block (8 waves) per (b,h,q-tile of 16 rows).
// Phase 1: scores strip [16x1024] via WMMA -> LDS (scaled, masked)
// Phase 2: per-row softmax (shfl reductions), write P to attn_weights + LDS
// Phase 3: ctx[16x64] = P @ V; all 8 waves: (dtile, k-half) split + LDS reduce
// Dynamic LDS: 16*A_STRIDE scores + 4*16*16 reduction = 69888 B.
// ---------------------------------------------------------------------------
__global__ void __launch_bounds__(256)
mha_attn(const float* __restrict__ Q, const float* __restrict__ K,
         const float* __restrict__ V, const unsigned char* __restrict__ mask,
         float* __restrict__ attnW, float* __restrict__ ctx)
{
    extern __shared__ float sc[];             // [16][A_STRIDE] + red[4][256]
    float* red = sc + 16 * A_STRIDE;

    const int blk = blockIdx.x;               // 0..4095
    const int qt  = blk & 63;
    const int bh  = blk >> 6;                 // b*16 + h
    const int b   = bh >> 4;
    const int h   = bh & 15;
    const int q0  = qt << 4;

    const int wave = threadIdx.x >> 5;
    const int lane = threadIdx.x & 31;
    const int lh   = lane >> 4;
    const int lm   = lane & 15;

    const float* Qb = Q + (size_t)bh * S_ * DK_;
    const float* Kb = K + (size_t)bh * S_ * DK_;
    const float* Vb = V + (size_t)bh * S_ * DK_;

    // ---- Phase 1: scores = (Q_tile @ K^T) / 8, masked -> LDS ----
    {
        const float* aRow = Qb + (size_t)(q0 + lm) * DK_;
        for (int nt = wave * 8; nt < wave * 8 + 8; ++nt) {
            const int c0 = nt << 4;
            const float* bRow = Kb + (size_t)(c0 + lm) * DK_;
            v8f acc = {0.f,0.f,0.f,0.f,0.f,0.f,0.f,0.f};
#pragma unroll
            for (int kk = 0; kk < DK_; kk += 4) {
                const int ks = kk + lh * 2;
                v2f a;  a.x  = aRow[ks]; a.y  = aRow[ks + 1];
                v2f bb; bb.x = bRow[ks]; bb.y = bRow[ks + 1];
                acc = wmma_f32(a, bb, acc);
            }
#pragma unroll
            for (int vv = 0; vv < 8; ++vv) {
                const int M = vv + 8 * lh;
                float s = acc[vv] * 0.125f;   // 1/sqrt(64)
                const unsigned char mk =
                    mask[((size_t)b * S_ + (q0 + M)) * S_ + c0 + lm];
                if (!mk) s = -INFINITY;
                sc[M * A_STRIDE + c0 + lm] = s;
            }
        }
    }
    __syncthreads();

    // ---- Phase 2: softmax per row; wave w owns rows 2w, 2w+1 ----
    for (int rr = 0; rr < 2; ++rr) {
        const int row = wave * 2 + rr;
        float* srow = sc + row * A_STRIDE;
        float mx = -INFINITY;
        for (int i = lane; i < S_; i += 32) mx = fmaxf(mx, srow[i]);
#pragma unroll
        for (int off = 16; off; off >>= 1)
            mx = fmaxf(mx, __shfl_xor(mx, off, 32));
        float sum = 0.f;
        for (int i = lane; i < S_; i += 32) {
            const float e = __expf(srow[i] - mx);
            srow[i] = e;
            sum += e;
        }
#pragma unroll
        for (int off = 16; off; off >>= 1) sum += __shfl_xor(sum, off, 32);
        const float inv = 1.0f / sum;
        float* grow = attnW + ((size_t)bh * S_ + (q0 + row)) * S_;
        for (int i = lane; i < S_; i += 32) {
            const float p = srow[i] * inv;
            srow[i] = p;                      // keep P in LDS for phase 3
            grow[i] = p;                      // attn_weights output
        }
    }
    __syncthreads();

    // ---- Phase 3: ctx = P @ V; wave -> (dtile = w&3, khalf = w>>2) ----
    {
        const int dtile = wave & 3;
        const int khalf = wave >> 2;
        const int d0 = dtile << 4;
        const int k0 = khalf * (S_ / 2);
        v8f acc = {0.f,0.f,0.f,0.f,0.f,0.f,0.f,0.f};
        const float* aL = sc + lm * A_STRIDE;
#pragma unroll 4
        for (int kk = k0; kk < k0 + S_ / 2; kk += 4) {
            const int ks = kk + lh * 2;
            v2f a;  a.x  = aL[ks];                          // P[M=lm, ks]
                    a.y  = aL[ks + 1];
            v2f bb; bb.x = Vb[(size_t)ks * DK_ + d0 + lm];  // V[ks, N=lm]
                    bb.y = Vb[(size_t)(ks + 1) * DK_ + d0 + lm];
            acc = wmma_f32(a, bb, acc);
        }
        if (khalf == 1) {
#pragma unroll
            for (int vv = 0; vv < 8; ++vv)
                red[dtile * 256 + vv * 32 + lane] = acc[vv];
        }
        __syncthreads();
        if (khalf == 0) {
#pragma unroll
            for (int vv = 0; vv < 8; ++vv) {
                const int M = vv + 8 * lh;
                const int s = q0 + M;
                const float val = acc[vv] + red[dtile * 256 + vv * 32 + lane];
                ctx[((size_t)b * S_ + s) * D_ + h * DK_ + d0 + lm] = val;
            }
        }
    }
}

// ---------------------------------------------------------------------------
// Kernel 3: output projection. out[r,n] = sum_k ctx[r,k]*Wo[n,k] + bo[n]
// Same LDS-staged structure as kernel 1; flat [B,S,D] output.
// ---------------------------------------------------------------------------
__global__ void __launch_bounds__(256)
mha_out_proj(const float* __restrict__ x, const float* __restrict__ W,
             const float* __restrict__ bias, float* __restrict__ out)
{
    extern __shared__ float lds[];            // [16][A_STRIDE]

    const int mt = blockIdx.x;                // 0..255
    const int m0 = mt << 4;

    {
        const float4* src = (const float4*)(x + (size_t)m0 * D_);
        for (int f = threadIdx.x; f < 16 * (D_ / 4); f += 256) {
            const int row = f >> 8;
            const int c4  = f & 255;
            *(float4*)&lds[row * A_STRIDE + c4 * 4] = src[row * 256 + c4];
        }
    }
    __syncthreads();

    const int wave = threadIdx.x >> 5;
    const int lane = threadIdx.x & 31;
    const int lh   = lane >> 4;
    const int lm   = lane & 15;
    const float* aL = lds + lm * A_STRIDE;

    for (int p = 0; p < 2; ++p) {
        const int ng = p * 8 + wave;
        const int n0 = ng << 6;
        const float* bRow0 = W + (size_t)(n0 + lm) * D_;
        const float* bRow1 = bRow0 + (size_t)16 * D_;
        const float* bRow2 = bRow0 + (size_t)32 * D_;
        const float* bRow3 = bRow0 + (size_t)48 * D_;
        __builtin_prefetch(bRow0, 0, 0);
        __builtin_prefetch(bRow1, 0, 0);
        __builtin_prefetch(bRow2, 0, 0);
        __builtin_prefetch(bRow3, 0, 0);

        v8f acc[4];
        acc[0] = acc[1] = acc[2] = acc[3] = (v8f){0.f,0.f,0.f,0.f,0.f,0.f,0.f,0.f};

#pragma unroll 4
        for (int kk = 0; kk < D_; kk += 4) {
            const int ks = kk + lh * 2;
            v2f a;  a.x  = aL[ks];      a.y  = aL[ks + 1];
            v2f b0; b0.x = bRow0[ks];   b0.y = bRow0[ks + 1];
            v2f b1; b1.x = bRow1[ks];   b1.y = bRow1[ks + 1];
            v2f b2; b2.x = bRow2[ks];   b2.y = bRow2[ks + 1];
            v2f b3; b3.x = bRow3[ks];   b3.y = bRow3[ks + 1];
            acc[0] = wmma_f32(a, b0, acc[0]);
            acc[1] = wmma_f32(a, b1, acc[1]);
            acc[2] = wmma_f32(a, b2, acc[2]);
            acc[3] = wmma_f32(a, b3, acc[3]);
        }

#pragma unroll
        for (int j = 0; j < 4; ++j) {
            const float bval = bias[n0 + j * 16 + lm];
#pragma unroll
            for (int vv = 0; vv < 8; ++vv) {
                const int r = m0 + vv + 8 * lh;
                out[(size_t)r * D_ + n0 + j * 16 + lm] = acc[j][vv] + bval;
            }
        }
    }
}

// ---------------------------------------------------------------------------
extern "C" void kernel_launch(void* const* d_in, const int* in_sizes, int n_in,
                              void* d_out, int out_size, void* d_ws, size_t ws_size,
                              hipStream_t stream) {
    (void)in_sizes; (void)n_in; (void)out_size; (void)ws_size;

    const float* q    = (const float*)d_in[0];
    const float* k    = (const float*)d_in[1];
    const float* v    = (const float*)d_in[2];
    const unsigned char* mask = (const unsigned char*)d_in[3];
    const float* Wq   = (const float*)d_in[4];
    const float* bq   = (const float*)d_in[5];
    const float* Wk   = (const float*)d_in[6];
    const float* bk   = (const float*)d_in[7];
    const float* Wv   = (const float*)d_in[8];
    const float* bv   = (const float*)d_in[9];
    const float* Wo   = (const float*)d_in[10];
    const float* bo   = (const float*)d_in[11];

    const size_t QKV_ELEMS = (size_t)B_ * H_ * S_ * DK_;   // 4M floats
    float* Qws = (float*)d_ws;
    float* Kws = Qws + QKV_ELEMS;
    float* Vws = Kws + QKV_ELEMS;
    float* ctx = Vws + QKV_ELEMS;

    float* outp  = (float*)d_out;                          // [B,S,D]
    float* attnW = outp + (size_t)B_ * S_ * D_;            // [B,H,S,S]

    const size_t stripBytes = (size_t)16 * A_STRIDE * sizeof(float);  // 65792
    dim3 blk(256);

    // 1) QKV projections: one block per (matrix, 16-row strip).
    mha_qkv_proj<<<dim3(256, 3), blk, stripBytes, stream>>>(
        q, k, v, Wq, Wk, Wv, bq, bk, bv, Qws, Kws, Vws);

    // 2) Attention core: one block per (b,h,q-tile16) = 4096 blocks.
    const size_t attnShm = stripBytes + 4 * 256 * sizeof(float);      // 69888
    mha_attn<<<dim3(4096), blk, attnShm, stream>>>(Qws, Kws, Vws, mask,
                                                   attnW, ctx);

    // 3) Output projection.
    mha_out_proj<<<dim3(256), blk, stripBytes, stream>>>(ctx, Wo, bo, outp);
}